// PEPS_NN_Model_1391569404359
// MI455X (gfx1250) — compile-verified
//
#include <hip/hip_runtime.h>
#include <hip/hip_bf16.h>

typedef __attribute__((ext_vector_type(2))) float v2f;
typedef __attribute__((ext_vector_type(8))) float v8f;
typedef __attribute__((ext_vector_type(4))) unsigned int u32x4;
typedef __attribute__((ext_vector_type(8))) int i32x8;
typedef __attribute__((ext_vector_type(4))) int i32x4;

static constexpr int   LOUT   = 197632;     // total env vector length
static constexpr int   BOTSZ  = 98816;      // bottom half size (= LOUT/2)
static constexpr float ETA    = 1e-3f;

__device__ __forceinline__ int offB(int px) { return (px == 0) ? 0 : 256 + (px - 1) * 16384; }

__device__ __forceinline__ v8f wmma_f32(v2f a, v2f b, v8f c) {
    return __builtin_amdgcn_wmma_f32_16x16x4_f32(false, a, false, b, (short)0, c, false, false);
}

// ---------------------------------------------------------------------------
// TDM: DMA one contiguous row of `nelem` f32 from global memory into LDS at
// byte offset `lds_off`.  D# per CDNA5 ISA ch.8: group0 = {count=1, lds_addr,
// global_addr[56:0], type=2}; group1 = {data_size=4B, tensor_dim0 = tile_dim0
// = nelem, tensor_dim1 = tile_dim1 = 1}.  Tracked by TENSORcnt.
// (clang-23 toolchain: 6-arg builtin with trailing i32x8 group + cpol)
// ---------------------------------------------------------------------------
__device__ __forceinline__ void tdm_load_row(unsigned lds_off, const void* gptr,
                                             unsigned nelem) {
    unsigned long long ga = (unsigned long long)(uintptr_t)gptr;
    u32x4 g0;
    g0[0] = 1u;                                              // count=1 (user D#)
    g0[1] = lds_off;                                         // lds_addr (bytes)
    g0[2] = (unsigned)(ga & 0xffffffffu);                    // global_addr[31:0]
    g0[3] = (unsigned)((ga >> 32) & 0x01ffffffu) | (2u << 30); // addr[56:32] | type=2
    i32x8 g1;
    g1[0] = (2 << 16);                                       // data_size=2 -> 4 bytes
    g1[1] = (int)((nelem & 0xffffu) << 16);                  // tensor_dim0[15:0]
    g1[2] = (int)(((nelem >> 16) & 0xffffu) | (1u << 16));   // dim0[31:16] | tensor_dim1=1
    g1[3] = (int)((nelem & 0xffffu) << 16);                  // tile_dim0
    g1[4] = 1;                                               // tile_dim1=1, tile_dim2=0
    g1[5] = (int)nelem;                                      // tensor_dim0_stride[31:0]
    g1[6] = 0;                                               // stride hi | dim1_stride lo
    g1[7] = 0;
    i32x4 z4 = {0, 0, 0, 0};
    i32x8 z8 = {0, 0, 0, 0, 0, 0, 0, 0};
    __builtin_amdgcn_tensor_load_to_lds(g0, g1, z4, z4, z8, 0);
}

// ---------------------------------------------------------------------------
// Kernel 1: H[s][k] = relu(x[s]@W1 + b1)   (256 x 64, trivial)
// ---------------------------------------------------------------------------
__global__ __launch_bounds__(256) void k_hidden(const int* __restrict__ x,
                                                const float* __restrict__ w1,
                                                const float* __restrict__ b1,
                                                float* __restrict__ H) {
    int s = threadIdx.x;                 // one sample per thread, grid = 1 block
    float xi[48];
#pragma unroll
    for (int i = 0; i < 48; ++i) xi[i] = (float)x[s * 48 + i];
    for (int k = 0; k < 64; ++k) {
        float acc = b1[k];
#pragma unroll
        for (int i = 0; i < 48; ++i) acc += xi[i] * w1[i * 64 + k];
        H[s * 64 + k] = fmaxf(acc, 0.0f);
    }
}

// ---------------------------------------------------------------------------
// Kernel 2: env build. One block per sample. Builds bottom/top three-row
// transfer tensors per column px and writes them in reference ravel order.
// ---------------------------------------------------------------------------
__global__ __launch_bounds__(256) void k_env(const float* __restrict__ peps,
                                             const int* __restrict__ x,
                                             float* __restrict__ env) {
    __shared__ float S[6][256];   // gathered site tensors (l,r,d,u) for one column
    __shared__ float M1[1024];    // intermediate (l01, r01, u)
    const int tid = threadIdx.x;
    const int b   = blockIdx.x;
    float* ev = env + (size_t)b * LOUT;

    for (int px = 0; px < 8; ++px) {
        const int ld  = (px == 0) ? 1 : 4;
        const int rd  = (px == 7) ? 1 : 4;
        const int rd2 = rd * rd;
        const int Lb  = ld * ld * ld;
        const int Rb  = rd * rd * rd;
        const int c1  = ld * ld * rd2 * 4;
        const int c2  = Lb * Rb * 4;

        __syncthreads();
        // gather 6 site tensors of this column at the sampled spin
        for (int i = tid; i < 6 * 256; i += 256) {
            int y = i >> 8, idx = i & 255;
            int spin = x[b * 48 + px * 6 + y];
            S[y][idx] = peps[(size_t)((px * 6 + y) * 256 + idx) * 2 + spin];
        }
        __syncthreads();

        // ---- bottom: rows 0,1 -> M1 ----
        for (int e = tid; e < c1; e += 256) {
            int u1 = e & 3, rest = e >> 2;
            int r01 = rest % rd2, l01 = rest / rd2;
            int l0 = l01 / ld, l1 = l01 % ld, r0 = r01 / rd, r1 = r01 % rd;
            float acc = 0.f;
#pragma unroll
            for (int u0 = 0; u0 < 4; ++u0)
                acc += S[0][(l0 * 4 + r0) * 16 + u0] * S[1][((l1 * 4 + r1) * 4 + u0) * 4 + u1];
            M1[e] = acc;
        }
        __syncthreads();
        // ---- bottom: M1 x row 2 -> env ----
        for (int e = tid; e < c2; e += 256) {
            int u2 = e & 3, rest = e >> 2;
            int r = rest % Rb, l = rest / Rb;
            int l01 = l / ld, l2 = l % ld, r01 = r / rd, r2 = r % rd;
            float acc = 0.f;
#pragma unroll
            for (int u1 = 0; u1 < 4; ++u1)
                acc += M1[(l01 * rd2 + r01) * 4 + u1] * S[2][((l2 * 4 + r2) * 4 + u1) * 4 + u2];
            ev[offB(px) + e] = acc;
        }
        __syncthreads();

        // ---- top: rows 5,4 -> M1 ----
        for (int e = tid; e < c1; e += 256) {
            int d4 = e & 3, rest = e >> 2;
            int r54 = rest % rd2, l54 = rest / rd2;
            int l5 = l54 / ld, l4 = l54 % ld, r5 = r54 / rd, r4 = r54 % rd;
            float acc = 0.f;
#pragma unroll
            for (int d5 = 0; d5 < 4; ++d5)
                acc += S[5][((l5 * 4 + r5) * 4 + d5) * 4] * S[4][((l4 * 4 + r4) * 4 + d4) * 4 + d5];
            M1[e] = acc;
        }
        __syncthreads();
        // ---- top: M1 x row 3 -> env ----
        for (int e = tid; e < c2; e += 256) {
            int d3 = e & 3, rest = e >> 2;
            int r = rest % Rb, l = rest / Rb;
            int l54 = l / ld, l3 = l % ld, r54 = r / rd, r3 = r % rd;
            float acc = 0.f;
#pragma unroll
            for (int d4 = 0; d4 < 4; ++d4)
                acc += M1[(l54 * rd2 + r54) * 4 + d4] * S[3][((l3 * 4 + r3) * 4 + d3) * 4 + d4];
            ev[BOTSZ + offB(px) + e] = acc;
        }
    }
}

// ---------------------------------------------------------------------------
// Kernel 3: env += ETA*(H @ W2 + b2).  WMMA GEMM  M=256, K=64, N=LOUT.
// One block = one 16-column stripe over all 256 rows; W2 read exactly once.
// ---------------------------------------------------------------------------
__global__ __launch_bounds__(256) void k_correct(const float* __restrict__ H,
                                                 const float* __restrict__ w2,
                                                 const float* __restrict__ b2,
                                                 float* __restrict__ env) {
    const int lane = threadIdx.x & 31;
    const int wave = threadIdx.x >> 5;
    const int half = lane >> 4;
    const int l16  = lane & 15;
    const int col0 = blockIdx.x * 16;

    // preload the 64x16 W2 panel as 16 B-fragments
    v2f bf[16];
#pragma unroll
    for (int kk = 0; kk < 16; ++kk) {
        int k0 = 4 * kk + (half ? 2 : 0);
        bf[kk].x = w2[(size_t)k0 * LOUT + col0 + l16];
        bf[kk].y = w2[(size_t)(k0 + 1) * LOUT + col0 + l16];
    }
    const float bias = b2[col0 + l16];

#pragma unroll
    for (int sub = 0; sub < 2; ++sub) {
        const int mbase = wave * 32 + sub * 16;
        v8f acc = {0.f, 0.f, 0.f, 0.f, 0.f, 0.f, 0.f, 0.f};
#pragma unroll
        for (int kk = 0; kk < 16; ++kk) {
            int k0 = 4 * kk + (half ? 2 : 0);
            v2f a;
            a.x = H[(mbase + l16) * 64 + k0];
            a.y = H[(mbase + l16) * 64 + k0 + 1];
            acc = wmma_f32(a, bf[kk], acc);
        }
#pragma unroll
        for (int v = 0; v < 8; ++v) {
            size_t idx = (size_t)(mbase + v + half * 8) * LOUT + col0 + l16;
            env[idx] += ETA * (acc[v] + bias);
        }
    }
}

// ---------------------------------------------------------------------------
// Kernel 4: boundary chain contraction, one block per sample, all in LDS.
//   E'(r,s) = sum_{a,b,u} E(a,b) nb(a,r,u) nt(b,s,u)   via two WMMA GEMMs.
// NB/NT tiles are DMA'd into LDS by the Tensor Data Mover (TENSORcnt).
// LDS: E 16KB | NB 64KB | NT 64KB | T 64KB | red 1KB  = 214016 B
// ---------------------------------------------------------------------------
__global__ __launch_bounds__(256) void k_chain(const float* __restrict__ vec_all,
                                               float* __restrict__ out) {
    extern __shared__ float sm[];
    float* E   = sm;             // 4096
    float* NB  = sm + 4096;      // 16384
    float* NT  = sm + 20480;     // 16384
    float* T   = sm + 36864;     // 16384
    float* red = sm + 53248;     // 256
    const int tid  = threadIdx.x;
    const int lane = tid & 31, wave = tid >> 5;
    const int half = lane >> 4, l16 = lane & 15;
    const float* vec = vec_all + (size_t)blockIdx.x * LOUT;
    const unsigned ldsbase = __builtin_amdgcn_groupstaticsize();  // dynamic-LDS base

    // ---- px = 0 (edge, rank-1): E(r,s) = sum_u nb0(r,u) nt0(s,u) ----
    NB[tid] = vec[tid];
    NT[tid] = vec[BOTSZ + tid];
    __syncthreads();
    for (int e = tid; e < 4096; e += 256) {
        int r = e >> 6, s = e & 63;
        float a = 0.f;
#pragma unroll
        for (int u = 0; u < 4; ++u) a += NB[r * 4 + u] * NT[s * 4 + u];
        E[e] = a;
    }
    __syncthreads();

    // ---- px = 1..6: two WMMA GEMMs per column ----
    for (int px = 1; px < 7; ++px) {
        const int ob = offB(px), ot = BOTSZ + ob;
        // TDM DMA: wave0 -> NB tile, wave1 -> NT tile (EXEC-independent, one
        // issue per wave; other waves see the data after wait + barrier).
        if (wave == 0)      tdm_load_row(ldsbase + 4096u * 4u,  vec + ob, 16384u);
        else if (wave == 1) tdm_load_row(ldsbase + 20480u * 4u, vec + ot, 16384u);
        __builtin_amdgcn_s_wait_tensorcnt(0);
        __syncthreads();

        // GEMM1: T(b, j) = sum_a E(a,b) * NB(a, j)   [64 x 256, K=64]
        for (int t8 = 0; t8 < 8; ++t8) {
            const int tile = wave * 8 + t8;
            const int bt = tile >> 4, jt = tile & 15;
            v8f acc = {0.f, 0.f, 0.f, 0.f, 0.f, 0.f, 0.f, 0.f};
#pragma unroll
            for (int kk = 0; kk < 16; ++kk) {
                int k0 = 4 * kk + (half ? 2 : 0);
                v2f a, bb;
                a.x  = E[k0 * 64 + bt * 16 + l16];          // A(m,k) = E(k, b)
                a.y  = E[(k0 + 1) * 64 + bt * 16 + l16];
                bb.x = NB[k0 * 256 + jt * 16 + l16];
                bb.y = NB[(k0 + 1) * 256 + jt * 16 + l16];
                acc  = wmma_f32(a, bb, acc);
            }
#pragma unroll
            for (int v = 0; v < 8; ++v)
                T[(bt * 16 + v + half * 8) * 256 + jt * 16 + l16] = acc[v];
        }
        __syncthreads();

        // GEMM2: E'(r,s) = sum_{p=(b,u)} T(b, r*4+u) * NT(b, s*4+u)  [64x64, K=256]
        v8f eo[2];
        for (int t2 = 0; t2 < 2; ++t2) {
            const int tile = wave * 2 + t2;
            const int rt = tile >> 2, st = tile & 3;
            v8f acc = {0.f, 0.f, 0.f, 0.f, 0.f, 0.f, 0.f, 0.f};
            const int sel = half ? 2 : 0;
            const int r  = rt * 16 + l16;
            const int ss = st * 16 + l16;
            for (int kk = 0; kk < 64; ++kk) {
                v2f a, bb;
                a.x  = T[kk * 256 + r * 4 + sel];
                a.y  = T[kk * 256 + r * 4 + sel + 1];
                bb.x = NT[kk * 256 + ss * 4 + sel];
                bb.y = NT[kk * 256 + ss * 4 + sel + 1];
                acc  = wmma_f32(a, bb, acc);
            }
            eo[t2] = acc;
        }
        __syncthreads();
        for (int t2 = 0; t2 < 2; ++t2) {
            const int tile = wave * 2 + t2;
            const int rt = tile >> 2, st = tile & 3;
#pragma unroll
            for (int v = 0; v < 8; ++v)
                E[(rt * 16 + v + half * 8) * 64 + st * 16 + l16] = eo[t2][v];
        }
        __syncthreads();
    }

    // ---- px = 7 (edge): scalar = sum_{a,b,u} E(a,b) nb7(a,u) nt7(b,u) ----
    NB[tid] = vec[offB(7) + tid];
    NT[tid] = vec[BOTSZ + offB(7) + tid];
    __syncthreads();
    {
        int a = tid >> 2, u = tid & 3;
        float acc = 0.f;
        for (int b = 0; b < 64; ++b) acc += E[a * 64 + b] * NT[b * 4 + u];
        red[tid] = acc * NB[tid];
    }
    __syncthreads();
    for (int s = 128; s > 0; s >>= 1) {
        if (tid < s) red[tid] += red[tid + s];
        __syncthreads();
    }
    if (tid == 0) out[blockIdx.x] = red[0];
}

// ---------------------------------------------------------------------------
// Host launcher.  d_in order: peps, w1, b1, w2, b2, x.
// ws layout: [0,64KB) H ; [64KB, 64KB+256*LOUT*4) env/vec (updated in place).
// Requires ws_size >= ~193.2 MB.
// ---------------------------------------------------------------------------
extern "C" void kernel_launch(void* const* d_in, const int* in_sizes, int n_in,
                              void* d_out, int out_size, void* d_ws, size_t ws_size,
                              hipStream_t stream) {
    const float* peps = (const float*)d_in[0];
    const float* w1   = (const float*)d_in[1];
    const float* b1   = (const float*)d_in[2];
    const float* w2   = (const float*)d_in[3];
    const float* b2   = (const float*)d_in[4];
    const int*   x    = (const int*)d_in[5];
    float* out = (float*)d_out;

    float* H   = (float*)d_ws;
    float* env = (float*)((char*)d_ws + 65536);

    k_hidden <<<1, 256, 0, stream>>>(x, w1, b1, H);
    k_env    <<<256, 256, 0, stream>>>(peps, x, env);
    k_correct<<<LOUT / 16, 256, 0, stream>>>(H, w2, b2, env);
    k_chain  <<<256, 256, 214016, stream>>>(env, out);
}